// ST_BIFNeuron_MS_89859305767691
// MI455X (gfx1250) — compile-verified
//
#include <hip/hip_runtime.h>

// ST-BIF multi-step neuron, memory-bound streaming scan.
// x: [T, N] fp32 (flattened), out: [T, N] fp32, per-element state carried in
// registers across the T (=4) timesteps. One thread owns one float4 chunk of
// the N spatial elements across all timesteps.

typedef float f4 __attribute__((ext_vector_type(4)));

// Fast-path step (valid for T <= 15 and q > 0):
//  - tc < POS_MAX(15) is always true (tc <= step_index <= 3), so pos = H >= q.
//  - state tq = tc*q is exact (k*q, k in [0,4]), so (tq > 0) <=> (tc > 0).
//  - spike*q selected directly: sq in {q, -q, 0}; it is the output value,
//    the v correction, and the tq increment. Bit-identical to the reference.
__device__ __forceinline__ f4 stbif_step_fast(f4 xt, f4& v, f4& tq, float q) {
    f4 s;
#pragma unroll
    for (int k = 0; k < 4; ++k) {
        float H  = v[k] + xt[k];
        bool pos = (H >= q);
        bool neg = (H < 0.0f) && (tq[k] > 0.0f);
        float sq = pos ? q : (neg ? -q : 0.0f);
        v[k]  = H - sq;
        tq[k] += sq;
        s[k]  = sq;
    }
    return s;
}

// Full-fidelity scalar step (any T, any q): exact reference semantics.
__device__ __forceinline__ float stbif_step_ref(float xt, float& v, float& tc, float q) {
    float H  = v + xt;
    bool pos = (H >= q) && (tc < 15.0f);
    bool neg = (H < 0.0f) && (tc > 0.0f);
    float sp = pos ? 1.0f : (neg ? -1.0f : 0.0f);
    float sq = sp * q;
    v  = H - sq;
    tc += sp;
    return sq;
}

__global__ void __launch_bounds__(256) stbif_ms_kernel(
    const float* __restrict__ x,
    const float* __restrict__ q_ptr,
    const int* __restrict__ nstep_ptr,
    float* __restrict__ out,
    unsigned int total)
{
    const float q = q_ptr[0];
    const int   T = nstep_ptr[0];
    const unsigned int N  = total / (unsigned int)T;  // elements per timestep
    const unsigned int N4 = N >> 2;                   // float4 chunks per timestep

    const f4* __restrict__ xv = (const f4*)x;
    f4*       __restrict__ ov = (f4*)out;

    const unsigned int tid    = blockIdx.x * blockDim.x + threadIdx.x;
    const unsigned int stride = gridDim.x * blockDim.x;

    if (T == 4 && q > 0.0f) {
        // Fast path: all 4 timestep loads in flight before first use,
        // minimal VALU per step, 32-bit indexing throughout.
        for (unsigned int i = tid; i < N4; i += stride) {
            f4 x0 = __builtin_nontemporal_load(&xv[i]);
            f4 x1 = __builtin_nontemporal_load(&xv[N4 + i]);
            f4 x2 = __builtin_nontemporal_load(&xv[2u * N4 + i]);
            f4 x3 = __builtin_nontemporal_load(&xv[3u * N4 + i]);

            f4 v, tq, s0;
            // Step 0 degenerates: tc==0 => neg impossible; pos = H >= q.
#pragma unroll
            for (int k = 0; k < 4; ++k) {
                float H  = 0.5f * q + x0[k];
                float sq = (H >= q) ? q : 0.0f;
                v[k]  = H - sq;
                tq[k] = sq;
                s0[k] = sq;
            }
            __builtin_nontemporal_store(s0, &ov[i]);

            f4 s1 = stbif_step_fast(x1, v, tq, q);
            __builtin_nontemporal_store(s1, &ov[N4 + i]);
            f4 s2 = stbif_step_fast(x2, v, tq, q);
            __builtin_nontemporal_store(s2, &ov[2u * N4 + i]);
            f4 s3 = stbif_step_fast(x3, v, tq, q);
            __builtin_nontemporal_store(s3, &ov[3u * N4 + i]);
        }
    } else {
        // Generic path: exact reference semantics for any T / q.
        for (unsigned int i = tid; i < N4; i += stride) {
            float v0 = 0.5f * q, v1 = 0.5f * q, v2 = 0.5f * q, v3 = 0.5f * q;
            float c0 = 0.0f, c1 = 0.0f, c2 = 0.0f, c3 = 0.0f;
            for (int t = 0; t < T; ++t) {
                f4 xt = __builtin_nontemporal_load(&xv[(unsigned int)t * N4 + i]);
                f4 st;
                st[0] = stbif_step_ref(xt[0], v0, c0, q);
                st[1] = stbif_step_ref(xt[1], v1, c1, q);
                st[2] = stbif_step_ref(xt[2], v2, c2, q);
                st[3] = stbif_step_ref(xt[3], v3, c3, q);
                __builtin_nontemporal_store(st, &ov[(unsigned int)t * N4 + i]);
            }
        }
    }

    // Scalar tail if N is not a multiple of 4 (not hit for the reference
    // shapes, kept for correctness under any shape).
    const unsigned int tail_start = N4 << 2;
    for (unsigned int e = tail_start + tid; e < N; e += stride) {
        float v = 0.5f * q, tc = 0.0f;
        for (int t = 0; t < T; ++t) {
            float xt = x[(unsigned int)t * N + e];
            out[(unsigned int)t * N + e] = stbif_step_ref(xt, v, tc, q);
        }
    }
}

extern "C" void kernel_launch(void* const* d_in, const int* in_sizes, int n_in,
                              void* d_out, int out_size, void* d_ws, size_t ws_size,
                              hipStream_t stream) {
    const float* x     = (const float*)d_in[0];
    const float* q_thr = (const float*)d_in[1];
    const int*   nstep = (const int*)d_in[2];
    float*       out   = (float*)d_out;

    const unsigned int total = (unsigned int)in_sizes[0];

    // Grid sized for the known setup (T=4, float4 per thread => exactly one
    // chunk per thread). The kernel grid-strides, so correctness does not
    // depend on this guess — only fast-path efficiency does.
    const int block = 256;
    unsigned int threads_needed = total / 16u;  // total / (T_assumed=4 * vec4)
    if (threads_needed < 1u) threads_needed = 1u;
    unsigned int grid = (threads_needed + (unsigned int)block - 1u) / (unsigned int)block;

    stbif_ms_kernel<<<grid, block, 0, stream>>>(x, q_thr, nstep, out, total);
}